// HydrophobicContrastiveModel_85822036508803
// MI455X (gfx1250) — compile-verified
//
#include <hip/hip_runtime.h>
#include <math.h>
#include <stdint.h>

typedef __attribute__((ext_vector_type(2))) float v2f;
typedef __attribute__((ext_vector_type(8))) float v8f;

#define B_ 32
#define L_ 1024
#define H_ 1280
#define P_ 128
#define KSTEPS 320            /* H_/4 */
#define CHUNK_KS 32           /* K-steps staged per LDS chunk            */
#define NCHUNK 10             /* KSTEPS / CHUNK_KS                       */
#define CHUNK_BYTES 65536     /* CHUNK_KS * 2 halves * 128 cols * 8 B    */
#define INV_T 14.285714285714286f   /* 1/0.07 */
#define NEG_INF_ (-1.0e30f)

__device__ __forceinline__ v8f wmma_f32(v2f a, v2f b, v8f c) {
    // V_WMMA_F32_16X16X4_F32 : D = A(16x4,f32) * B(4x16,f32) + C(16x16,f32)
    return __builtin_amdgcn_wmma_f32_16x16x4_f32(
        false, a, false, b, (short)0, c, false, false);
}

// ---------------------------------------------------------------------------
// Kernel 0: pack proj_w into WMMA B-fragment order.
// Wf[(ks*2 + half)*128 + n] = { W[(4ks+2half)*128+n], W[(4ks+2half+1)*128+n] }
// so a lane's B fragment becomes a single coalesced 8-byte load.
// ---------------------------------------------------------------------------
__global__ __launch_bounds__(256) void k_packW(const float* __restrict__ W,
                                               v2f* __restrict__ Wf) {
    int idx  = blockIdx.x * 256 + threadIdx.x;   // 320*2*128 = 81920 fragments
    int ks   = idx >> 8;
    int half = (idx >> 7) & 1;
    int n    = idx & 127;
    int k    = 4 * ks + 2 * half;
    v2f f;
    f[0] = W[k * P_ + n];
    f[1] = W[(k + 1) * P_ + n];
    Wf[idx] = f;
}

// ---------------------------------------------------------------------------
// Kernel 1: emb = normalize(H @ W + bias).  One wave -> 16 rows x 128 cols.
// The 8 waves of a block share Wf via LDS, staged with async load-to-LDS.
// ---------------------------------------------------------------------------
__global__ __launch_bounds__(256) void k_proj(const float* __restrict__ H,
                                              const v2f* __restrict__ Wf,
                                              const float* __restrict__ bias,
                                              float* __restrict__ emb) {
    extern __shared__ char smem[];               // CHUNK_BYTES of staged Wf
    const v2f* sW = (const v2f*)smem;
    uint32_t sbase = (uint32_t)(uintptr_t)(void*)smem;

    int tid  = threadIdx.x;
    int wave = blockIdx.x * 8 + (tid >> 5);
    int lane = tid & 31;
    int half = lane >> 4;       // 0: K {0,1}, rows M=v ; 1: K {2,3}, rows M=v+8
    int l    = lane & 15;
    long r0  = (long)wave * 16;

    const float* pA = H + (r0 + l) * H_ + 2 * half;

    v8f zero = {};
    v8f acc[8];
#pragma unroll
    for (int t = 0; t < 8; ++t) acc[t] = zero;

    for (int chunk = 0; chunk < NCHUNK; ++chunk) {
        // --- cooperative async copy: 64 KB of Wf fragments -> LDS ---------
        const char* src = (const char*)(Wf + (size_t)chunk * (CHUNK_KS * 2 * P_));
#pragma unroll
        for (int i = 0; i < 16; ++i) {
            uint32_t off = (uint32_t)tid * 16u + (uint32_t)i * 4096u;
            uint32_t dst = sbase + off;
            asm volatile("global_load_async_to_lds_b128 %0, %1, %2"
                         :: "v"(dst), "v"(off), "s"(src) : "memory");
        }
        asm volatile("s_wait_asynccnt 0x0" ::: "memory");
        __syncthreads();

        // --- consume CHUNK_KS K-steps from LDS ----------------------------
        const float* pAc = pA + 4 * (chunk * CHUNK_KS);
#pragma unroll 8
        for (int ksl = 0; ksl < CHUNK_KS; ++ksl) {
            v2f a = *(const v2f*)(pAc + 4 * ksl);
            int fb = (ksl * 2 + half) * P_ + l;
#pragma unroll
            for (int t = 0; t < 8; ++t) {
                v2f b = sW[fb + t * 16];
                acc[t] = wmma_f32(a, b, acc[t]);
            }
        }
        __syncthreads();    // protect LDS before next chunk overwrites it
    }

    // + bias (per column)
#pragma unroll
    for (int t = 0; t < 8; ++t) {
        float bb = bias[t * 16 + l];
#pragma unroll
        for (int v = 0; v < 8; ++v) acc[t][v] += bb;
    }

    // L2 normalize each row: reduce sum-of-squares over the 16 lanes of a half
#pragma unroll
    for (int v = 0; v < 8; ++v) {
        float ss = 0.f;
#pragma unroll
        for (int t = 0; t < 8; ++t) ss += acc[t][v] * acc[t][v];
        ss += __shfl_xor(ss, 1, 32);
        ss += __shfl_xor(ss, 2, 32);
        ss += __shfl_xor(ss, 4, 32);
        ss += __shfl_xor(ss, 8, 32);
        float rn = 1.0f / sqrtf(ss);
#pragma unroll
        for (int t = 0; t < 8; ++t) acc[t][v] *= rn;
    }

    // store: row = r0 + v + 8*half, col = t*16 + l
#pragma unroll
    for (int v = 0; v < 8; ++v) {
        long row = r0 + v + 8 * half;
#pragma unroll
        for (int t = 0; t < 8; ++t)
            emb[row * P_ + t * 16 + l] = acc[t][v];
    }
}

// ---------------------------------------------------------------------------
// Kernel 2: fused sim = (E E^T)/T + online masked softmax statistics.
// One wave -> 16 anchor rows of one batch; loops 64 j-tiles x 32 K-step WMMAs.
// Writes contrib[b][i] = valid[i] * anchor_loss[i].
// ---------------------------------------------------------------------------
__global__ __launch_bounds__(256) void k_loss(const float* __restrict__ emb,
                                              const int* __restrict__ attn,
                                              const int* __restrict__ labels,
                                              float* __restrict__ contrib) {
    int wave = blockIdx.x * 8 + (threadIdx.x >> 5);
    int lane = threadIdx.x & 31;
    int half = lane >> 4;
    int l    = lane & 15;
    int b    = wave >> 6;           // 32 batches
    int i0   = (wave & 63) * 16;    // 64 row tiles per batch

    const float* E  = emb + (long)b * L_ * P_;
    const int*   at = attn + b * L_;
    const int*   lb = labels + b * L_;

    // Cache anchor-tile A fragments (16 rows x 128 K) in registers.
    v2f af[32];
    {
        const float* pA = E + (i0 + l) * P_ + 2 * half;
#pragma unroll
        for (int ks = 0; ks < 32; ++ks) af[ks] = *(const v2f*)(pA + 4 * ks);
    }

    // Row metadata for the 8 rows this lane's half owns.
    float li[8], vi[8];
#pragma unroll
    for (int v = 0; v < 8; ++v) {
        int row = i0 + v + 8 * half;
        int lab = lb[row];
        li[v] = (float)lab;
        vi[v] = (at[row] != 0 && lab != -100) ? 1.f : 0.f;
    }

    float m[8], den[8], ps[8], cnt[8];
#pragma unroll
    for (int v = 0; v < 8; ++v) { m[v] = NEG_INF_; den[v] = 0.f; ps[v] = 0.f; cnt[v] = 0.f; }

    for (int jt = 0; jt < 64; ++jt) {
        int jcol = jt * 16 + l;              // this lane's column
        int la_j = lb[jcol];
        bool cv  = (at[jcol] != 0) && (la_j != -100);   // col_valid[j]
        float lj = (float)la_j;

        v8f c = {};
        const float* pB = E + jcol * P_ + 2 * half;      // B[k][n] = E[j0+n][k]
#pragma unroll
        for (int ks = 0; ks < 32; ++ks) {
            v2f bf = *(const v2f*)(pB + 4 * ks);
            c = wmma_f32(af[ks], bf, c);
        }

#pragma unroll
        for (int v = 0; v < 8; ++v) {
            float s    = c[v] * INV_T;
            int   irow = i0 + v + 8 * half;
            bool  diag = (irow == jcol);
            float nm   = cv ? fmaxf(m[v], s) : m[v];     // max includes diagonal
            float add  = (cv && !diag) ? __expf(s - nm) : 0.f;
            den[v]     = den[v] * __expf(m[v] - nm) + add;
            m[v]       = nm;
            bool isp   = cv && !diag && (lj == li[v]);
            if (isp) { ps[v] += s; cnt[v] += 1.f; }
        }
    }

    // Combine across the 16 lanes of each half (flash-style (m,den) merge).
#pragma unroll
    for (int v = 0; v < 8; ++v) {
#pragma unroll
        for (int d = 1; d < 16; d <<= 1) {
            float om = __shfl_xor(m[v], d, 32);
            float od = __shfl_xor(den[v], d, 32);
            float nm = fmaxf(m[v], om);
            den[v]   = den[v] * __expf(m[v] - nm) + od * __expf(om - nm);
            m[v]     = nm;
            ps[v]  += __shfl_xor(ps[v], d, 32);
            cnt[v] += __shfl_xor(cnt[v], d, 32);
        }
    }

    if (l == 0) {
#pragma unroll
        for (int v = 0; v < 8; ++v) {
            int row = i0 + v + 8 * half;
            float out = 0.f;
            if (vi[v] != 0.f) {
                // sum_pos(logits) - n_pos*log(den+eps), normalized by n_pos
                out = (ps[v] - cnt[v] * m[v] - cnt[v] * __logf(den[v] + 1e-12f))
                      / (cnt[v] + 1e-12f);
            }
            contrib[b * L_ + row] = out;
        }
    }
}

// ---------------------------------------------------------------------------
// Kernel 3: per-batch reduction -> (seq loss, seq_ok)
// ---------------------------------------------------------------------------
__global__ __launch_bounds__(256) void k_batch(const float* __restrict__ contrib,
                                               const int* __restrict__ attn,
                                               const int* __restrict__ labels,
                                               float* __restrict__ bl,
                                               float* __restrict__ bok) {
    int b = blockIdx.x;
    int tid = threadIdx.x;
    __shared__ float sC[256], sV[256], sA[256];
    float c = 0.f, nv = 0.f, na = 0.f;
    for (int j = tid; j < L_; j += 256) {
        int a   = attn[b * L_ + j];
        int lab = labels[b * L_ + j];
        c  += contrib[b * L_ + j];
        nv += (a != 0 && lab != -100) ? 1.f : 0.f;
        na += (float)a;
    }
    sC[tid] = c; sV[tid] = nv; sA[tid] = na;
    __syncthreads();
    for (int s = 128; s > 0; s >>= 1) {
        if (tid < s) { sC[tid] += sC[tid + s]; sV[tid] += sV[tid + s]; sA[tid] += sA[tid + s]; }
        __syncthreads();
    }
    if (tid == 0) {
        float n_anchor = fmaxf(sV[0], 1.f);
        float loss_seq = -sC[0] / n_anchor;
        bool ok = (sA[0] >= 2.f);
        bl[b]  = ok ? loss_seq : 0.f;
        bok[b] = ok ? 1.f : 0.f;
    }
}

// ---------------------------------------------------------------------------
// Kernel 4: final scalar
// ---------------------------------------------------------------------------
__global__ void k_final(const float* __restrict__ bl,
                        const float* __restrict__ bok,
                        float* __restrict__ out) {
    int lane = threadIdx.x;
    float v = bl[lane], o = bok[lane];
#pragma unroll
    for (int d = 1; d < 32; d <<= 1) {
        v += __shfl_xor(v, d, 32);
        o += __shfl_xor(o, d, 32);
    }
    if (lane == 0) out[0] = v / fmaxf(o, 1.f);
}

extern "C" void kernel_launch(void* const* d_in, const int* in_sizes, int n_in,
                              void* d_out, int out_size, void* d_ws, size_t ws_size,
                              hipStream_t stream) {
    const float* H     = (const float*)d_in[0];
    const float* W     = (const float*)d_in[1];
    const float* bias  = (const float*)d_in[2];
    const int*   attn  = (const int*)d_in[3];
    const int*   labels= (const int*)d_in[4];
    float* out = (float*)d_out;

    char* ws = (char*)d_ws;
    float* emb     = (float*)ws;                                   // 32768*128 f32 = 16 MiB
    float* contrib = (float*)(ws + (size_t)B_ * L_ * P_ * sizeof(float));
    float* bl      = contrib + B_ * L_;
    float* bok     = bl + B_;
    v2f*   Wf      = (v2f*)(bok + B_ + 32);                        // 655 KiB packed W

    // Pack W into B-fragment order: 81920 fragments / 256 = 320 blocks
    k_packW<<<320, 256, 0, stream>>>(W, Wf);
    // 2048 waves for projection (32768 rows / 16), 8 waves/block, 64 KiB LDS
    k_proj <<<256, 256, CHUNK_BYTES, stream>>>(H, Wf, bias, emb);
    // 2048 waves for fused sim+loss (32 batches * 64 row tiles)
    k_loss <<<256, 256, 0, stream>>>(emb, attn, labels, contrib);
    k_batch<<<B_,  256, 0, stream>>>(contrib, attn, labels, bl, bok);
    k_final<<<1,    32, 0, stream>>>(bl, bok, out);
}